// IdxPool_72344429134229
// MI455X (gfx1250) — compile-verified
//
#include <hip/hip_runtime.h>

// 2x2 pixel-unshuffle: x[32,1,2048,2048] f32 -> out[32,4,1024,1024] f32
// out[b, 2p+q, i, j] = x[b, 0, 2i+p, 2j+q]
//
// Pure streaming permute: 1 GiB total traffic, ~46 us floor at 23.3 TB/s.
// Strategy: 128-bit nontemporal loads/stores, register deinterleave,
// perfect coalescing on both read and write sides.

typedef __attribute__((ext_vector_type(4))) float f32x4;

#define W_IN      2048u
#define H_IN      2048u
#define W_OUT     1024u          // W_IN/2
#define PLANE_OUT (1024u * 1024u) // one output channel plane (floats)
#define ROWS_PER_BLOCK 4u

__global__ __launch_bounds__(256) void unshuffle2x2_kernel(
    const float* __restrict__ x, float* __restrict__ out)
{
    // Block owns ROWS_PER_BLOCK consecutive input rows of one image.
    // 2048 rows/image is divisible by 4, so all 4 rows share the same image b.
    const unsigned row0 = blockIdx.x * ROWS_PER_BLOCK;      // flat input row
    const unsigned b    = row0 >> 11;                        // row0 / 2048
    const unsigned tid  = threadIdx.x;                       // 0..255

    // Thread t covers input columns [8t, 8t+8) -> output columns [4t, 4t+4)
    const float* src = x + (size_t)row0 * W_IN + (size_t)tid * 8u;
    float* const out_img = out + (size_t)b * 4u * PLANE_OUT + (size_t)tid * 4u;

#pragma unroll
    for (unsigned rr = 0; rr < ROWS_PER_BLOCK; ++rr) {
        const unsigned r = (row0 + rr) & (H_IN - 1u);  // row within image
        const unsigned i = r >> 1;                     // output row
        const unsigned p = r & 1u;                     // 0 -> ch {0,1}, 1 -> ch {2,3}

        const f32x4 v0 = __builtin_nontemporal_load(
            (const f32x4*)(src + (size_t)rr * W_IN));
        const f32x4 v1 = __builtin_nontemporal_load(
            (const f32x4*)(src + (size_t)rr * W_IN + 4u));

        const f32x4 ev = { v0.x, v0.z, v1.x, v1.z };   // columns 2j   -> ch 2p
        const f32x4 od = { v0.y, v0.w, v1.y, v1.w };   // columns 2j+1 -> ch 2p+1

        float* oe = out_img + (size_t)(2u * p) * PLANE_OUT + (size_t)i * W_OUT;
        __builtin_nontemporal_store(ev, (f32x4*)oe);
        __builtin_nontemporal_store(od, (f32x4*)(oe + PLANE_OUT));
    }
}

extern "C" void kernel_launch(void* const* d_in, const int* in_sizes, int n_in,
                              void* d_out, int out_size, void* d_ws, size_t ws_size,
                              hipStream_t stream)
{
    (void)in_sizes; (void)n_in; (void)out_size; (void)d_ws; (void)ws_size;

    const float* x   = (const float*)d_in[0];
    float*       out = (float*)d_out;

    // 32 images * 2048 input rows / 4 rows per block = 16384 blocks
    const unsigned total_rows = 32u * H_IN;
    const dim3 grid(total_rows / ROWS_PER_BLOCK);
    const dim3 block(256);

    unshuffle2x2_kernel<<<grid, block, 0, stream>>>(x, out);
}